// call_FocalSparseConv_33801392620148
// MI455X (gfx1250) — compile-verified
//
#include <hip/hip_runtime.h>

typedef float v2f __attribute__((ext_vector_type(2)));
typedef float v8f __attribute__((ext_vector_type(8)));

namespace {
constexpr int kB = 2, kZ = 21, kY = 200, kX = 200;
constexpr int kCin = 32, kCout = 64;
constexpr int kNPts = 80000;
constexpr int kNVox = kB * kZ * kY * kX;              // 1,680,000
constexpr long long kNOut = (long long)kNVox * kCout; // 107,520,000
constexpr int kMaxTiles = (kNVox + 15) / 16;          // 105,000
constexpr int kBpackFloats = 27 * 4 * 8 * 32 * 2;     // 55,296
constexpr float kEps = 1e-3f;
}

// ---------------------------------------------------------------- init
__global__ void init_kernel(int* __restrict__ grid_id, int* __restrict__ active,
                            int* __restrict__ counter) {
  const int i = blockIdx.x * blockDim.x + threadIdx.x;
  if (i < kNVox) { grid_id[i] = -1; active[i] = 0; }
  if (i == 0) counter[0] = 0;
}

__global__ void zero_out_kernel(float4* __restrict__ out) {
  const long long i = (long long)blockIdx.x * blockDim.x + threadIdx.x;
  if (i < kNOut / 4) out[i] = make_float4(0.f, 0.f, 0.f, 0.f);
}

// ------------------------------------------------------- scatter points
__global__ void scatter_kernel(const int* __restrict__ indices,
                               int* __restrict__ grid_id, int* __restrict__ active) {
  const int n = blockIdx.x * blockDim.x + threadIdx.x;
  if (n >= kNPts) return;
  const int b = indices[n * 4 + 0], z = indices[n * 4 + 1];
  const int y = indices[n * 4 + 2], x = indices[n * 4 + 3];
  const int lin = ((b * kZ + z) * kY + y) * kX + x;
  grid_id[lin] = n;   // indices are unique per batch
  active[lin] = 1;    // input points are always active
}

// ---------------- prepack W_conv into WMMA-B lane layout, fold BN ------
// B tile for V_WMMA_F32_16X16X4_F32 (4x16 f32, 2 VGPRs):
//   VGPR j, lane L<16 : (K = kk*4 + j,     N = nt*16 + L)
//   VGPR j, lane L>=16: (K = kk*4 + 2 + j, N = nt*16 + (L-16))
__global__ void prepack_kernel(const float* __restrict__ Wc,
                               const float* __restrict__ g, const float* __restrict__ be,
                               const float* __restrict__ mu, const float* __restrict__ var,
                               float* __restrict__ bpack, float* __restrict__ bnscale,
                               float* __restrict__ bnshift) {
  const int tid = blockIdx.x * blockDim.x + threadIdx.x;
  if (tid < 27 * 4 * 8 * 32) {
    const int lane = tid & 31, kk = (tid >> 5) & 7, nt = (tid >> 8) & 3, tap = tid >> 10;
    const int K0 = kk * 4 + ((lane >= 16) ? 2 : 0);
    const int N = nt * 16 + (lane & 15);
    v2f w;
    w[0] = Wc[(tap * 32 + K0) * 64 + N];
    w[1] = Wc[(tap * 32 + K0 + 1) * 64 + N];
    *(v2f*)(bpack + (long long)tid * 2) = w;
  } else if (tid < 27 * 4 * 8 * 32 + 64) {
    const int co = tid - 27 * 4 * 8 * 32;
    const float s = g[co] * rsqrtf(var[co] + kEps);
    bnscale[co] = s;
    bnshift[co] = be[co] - mu[co] * s;
  }
}

// ------------------- importance conv at points + active-mask scatter ---
// One wave32 per point; lane = importance channel (0..26 used).
// sigmoid(v) > 0.5 <=> v > 0 ; sigmoid(v) >= 0.5 <=> v >= 0.
__global__ void __launch_bounds__(128) imp_kernel(
    const float* __restrict__ feats, const int* __restrict__ indices,
    const float* __restrict__ Wimp, const int* __restrict__ grid_id,
    int* __restrict__ active) {
  const int pt = blockIdx.x * 4 + (threadIdx.x >> 5);
  if (pt >= kNPts) return;                   // wave-uniform
  const int lane = threadIdx.x & 31;
  const int b = indices[pt * 4 + 0], z = indices[pt * 4 + 1];
  const int y = indices[pt * 4 + 2], x = indices[pt * 4 + 3];

  float acc = 0.f;
  for (int tap = 0; tap < 27; ++tap) {
    const int nz = z + tap / 9 - 1;
    const int ny = y + (tap / 3) % 3 - 1;
    const int nx = x + tap % 3 - 1;
    int m = -1;
    if ((unsigned)nz < (unsigned)kZ && (unsigned)ny < (unsigned)kY &&
        (unsigned)nx < (unsigned)kX)
      m = grid_id[((b * kZ + nz) * kY + ny) * kX + nx];  // wave-uniform
    if (m >= 0 && lane < 27) {
      const float* f = feats + (long long)m * kCin;      // uniform -> scalar loads
      const float* w = Wimp + tap * kCin * 27 + lane;    // coalesced over lanes
      float a = acc;
#pragma unroll 8
      for (int ci = 0; ci < kCin; ++ci) a = fmaf(f[ci], w[ci * 27], a);
      acc = a;
    }
  }
  const float imp26 = __shfl(acc, 26, 32);   // voxel-importance channel
  const bool fore = imp26 > 0.f;
  if (lane < 26 && fore && acc >= 0.f) {
    const int tap = (lane < 13) ? lane : lane + 1;   // KERNEL_OFFSETS skip center
    const int cz = z + tap / 9 - 1;
    const int cy = y + (tap / 3) % 3 - 1;
    const int cx = x + tap % 3 - 1;
    // reference bounds check is strict-lower: c > 0
    if (cz > 0 && cy > 0 && cx > 0 && cz < kZ && cy < kY && cx < kX)
      active[((b * kZ + cz) * kY + cy) * kX + cx] = 1;  // benign race, writes 1
  }
}

// ----------------------- compact active voxels (wave-aggregated) -------
__global__ void compact_kernel(const int* __restrict__ active,
                               int* __restrict__ list, int* __restrict__ counter) {
  const int i = blockIdx.x * blockDim.x + threadIdx.x;
  const bool pred = (i < kNVox) && (active[i] != 0);
  const unsigned mask = (unsigned)__ballot(pred);      // wave32: low 32 bits
  const int lane = threadIdx.x & 31;
  const int cnt = __popc(mask);
  int base = 0;
  if (lane == 0 && cnt) base = atomicAdd(counter, cnt);
  base = __shfl(base, 0, 32);
  if (pred) list[base + __popc(mask & ((1u << lane) - 1u))] = i;
}

// --------------- main sparse gather-conv: f32 WMMA 16x16x4 -------------
// One wave32 per 16-voxel tile. A (16x4 f32): VGPR j, lane L<16 holds
// (M=L, K=j); lane L>=16 holds (M=L-16, K=2+j). C/D: VGPR r, lanes 0-15
// = (M=r, N=lane), lanes 16-31 = (M=r+8, N=lane-16).
__global__ void __launch_bounds__(32) conv_kernel(
    const float* __restrict__ feats, const int* __restrict__ grid_id,
    const int* __restrict__ list, const int* __restrict__ counter,
    const float* __restrict__ bpack, const float* __restrict__ bnscale,
    const float* __restrict__ bnshift, float* __restrict__ out) {
  const int count = counter[0];
  const int tile = blockIdx.x;
  if (tile * 16 >= count) return;            // wave-uniform exit
  const int lane = threadIdx.x;
  const int m = lane & 15;
  const int gi = tile * 16 + m;
  const bool rowValid = gi < count;
  const int lin = rowValid ? list[gi] : 0;
  int t = lin;
  const int x = t % kX; t /= kX;
  const int y = t % kY; t /= kY;
  const int z = t % kZ;
  const int b = t / kZ;
  const int koff = (lane >= 16) ? 2 : 0;

  v8f acc[4] = {};

  for (int tap = 0; tap < 27; ++tap) {
    const int nz = z + tap / 9 - 1;
    const int ny = y + (tap / 3) % 3 - 1;
    const int nx = x + tap % 3 - 1;
    const bool inb = rowValid && (unsigned)nz < (unsigned)kZ &&
                     (unsigned)ny < (unsigned)kY && (unsigned)nx < (unsigned)kX;
    int row = -1;
    if (inb) row = grid_id[((b * kZ + nz) * kY + ny) * kX + nx];
    if (__ballot(row >= 0) == 0ull) continue;  // uniform skip of empty taps
    const float* fp = feats + (long long)row * kCin;
    const float* bp = bpack + (long long)tap * (4 * 8 * 64);
#pragma unroll
    for (int kk = 0; kk < 8; ++kk) {
      v2f a = {};                              // zero rows with no neighbor
      if (row >= 0) a = *(const v2f*)(fp + kk * 4 + koff);
#pragma unroll
      for (int nt = 0; nt < 4; ++nt) {
        const v2f bv = *(const v2f*)(bp + ((nt * 8 + kk) * 32 + lane) * 2);
        acc[nt] = __builtin_amdgcn_wmma_f32_16x16x4_f32(
            false, a, false, bv, (short)0, acc[nt], false, false);
      }
    }
  }

  // BN (prefolded scale/shift) + ReLU + scatter store
  const int hi = (lane >= 16) ? 8 : 0;
#pragma unroll
  for (int nt = 0; nt < 4; ++nt) {
    const int co = nt * 16 + (lane & 15);
    const float s = bnscale[co];
    const float sh = bnshift[co];
#pragma unroll
    for (int r = 0; r < 8; ++r) {
      const int mr = r + hi;
      const float v = fmaxf(acc[nt][r] * s + sh, 0.f);
      const int linr = __shfl(lin, mr, 32);
      if (tile * 16 + mr < count)
        out[(long long)linr * kCout + co] = v;
    }
  }
}

// ---------------------------------------------------------------- launch
extern "C" void kernel_launch(void* const* d_in, const int* in_sizes, int n_in,
                              void* d_out, int out_size, void* d_ws, size_t ws_size,
                              hipStream_t stream) {
  (void)in_sizes; (void)n_in; (void)out_size; (void)ws_size;
  const float* feats   = (const float*)d_in[0];
  const int*   indices = (const int*)d_in[1];
  const float* Wimp    = (const float*)d_in[2];
  const float* Wconv   = (const float*)d_in[3];
  const float* gamma   = (const float*)d_in[4];
  const float* beta    = (const float*)d_in[5];
  const float* mean    = (const float*)d_in[6];
  const float* var     = (const float*)d_in[7];
  float* out = (float*)d_out;

  char* ws = (char*)d_ws;                       // ~20.4 MB used
  int* grid_id = (int*)ws;                      // NVOX i32
  int* active  = grid_id + kNVox;               // NVOX i32
  int* list    = active + kNVox;                // NVOX i32
  int* counter = list + kNVox;                  // 1 i32 (+pad)
  float* bpack   = (float*)(ws + (size_t)3 * kNVox * 4 + 256);  // 8B aligned
  float* bnscale = bpack + kBpackFloats;
  float* bnshift = bnscale + 64;

  init_kernel<<<(kNVox + 255) / 256, 256, 0, stream>>>(grid_id, active, counter);
  zero_out_kernel<<<(int)(kNOut / 4 / 256), 256, 0, stream>>>((float4*)out);
  scatter_kernel<<<(kNPts + 255) / 256, 256, 0, stream>>>(indices, grid_id, active);
  prepack_kernel<<<(27 * 4 * 8 * 32 + 64 + 255) / 256, 256, 0, stream>>>(
      Wconv, gamma, beta, mean, var, bpack, bnscale, bnshift);
  imp_kernel<<<(kNPts + 3) / 4, 128, 0, stream>>>(feats, indices, Wimp, grid_id, active);
  compact_kernel<<<(kNVox + 255) / 256, 256, 0, stream>>>(active, list, counter);
  conv_kernel<<<kMaxTiles, 32, 0, stream>>>(feats, grid_id, list, counter,
                                            bpack, bnscale, bnshift, out);
}